// OptionsCritic_5892695130342
// MI455X (gfx1250) — compile-verified
//
#include <hip/hip_runtime.h>

typedef float v2f __attribute__((ext_vector_type(2)));
typedef float v4f __attribute__((ext_vector_type(4)));
typedef float v8f __attribute__((ext_vector_type(8)));

#define BATCH   32768
#define OBSD    256
#define HID     64
#define FEAT    128
#define NOPT    8
#define ACTN    18

// ---- workspace layout (float offsets) ----
#define OFF_STATE   0u                         // BATCH*FEAT            = 4194304
#define OFF_FW1P    4194304u                   // 4*64*64   fragments   = 16384
#define OFF_FW2P    4210688u                   // 8*16*64               = 8192
#define OFF_HEADP   4218880u                   // 1*32*64               = 2048
#define OFF_OW1P    4220928u                   // 8*4*32*64             = 65536
#define OFF_OW2P    4286464u                   // 8*2*16*64             = 16384
#define OFF_INTS    4302848u                   // 32 ints (counts/bases/cursors)
#define OFF_IDX     4302880u                   // BATCH ints

__device__ __forceinline__ void lds_fence() {
    // wave-private LDS handoff: all DS stores complete before dependent reads
    asm volatile("s_wait_dscnt 0" ::: "memory");
}

__device__ __forceinline__ v8f wmma4(v2f a, v2f b, v8f c) {
    return __builtin_amdgcn_wmma_f32_16x16x4_f32(false, a, false, b, (short)0, c,
                                                 false, false);
}

// ------------------------------------------------------------------
// Repack weights into per-lane WMMA B-fragment order:
//   frag(nt,kt): lane L holds B[k][n] for n = nt*16 + (L&15),
//                k = kt*4 + 2*(L>>4) + c, c in {0,1}  (B = W^T, so B[k][n]=W[n][k])
// ------------------------------------------------------------------
__global__ void pack_weights_kernel(const float* __restrict__ fW1,
                                    const float* __restrict__ fW2,
                                    const float* __restrict__ pW,
                                    const float* __restrict__ tW,
                                    const float* __restrict__ oW1,
                                    const float* __restrict__ oW2,
                                    float* __restrict__ ws) {
    int t = blockIdx.x * blockDim.x + threadIdx.x;
    if (t >= 108544) return;
    if (t < 16384) {                                   // fW1: [64,256] -> nt(4),kt(64)
        int c = t & 1, lane = (t >> 1) & 31, rest = t >> 6;
        int kt = rest & 63, nt = rest >> 6;
        int n = nt * 16 + (lane & 15);
        int k = kt * 4 + ((lane >> 4) << 1) + c;
        ws[OFF_FW1P + t] = fW1[n * OBSD + k];
    } else if (t < 24576) {                            // fW2: [128,64] -> nt(8),kt(16)
        int u = t - 16384;
        int c = u & 1, lane = (u >> 1) & 31, rest = u >> 6;
        int kt = rest & 15, nt = rest >> 4;
        int n = nt * 16 + (lane & 15);
        int k = kt * 4 + ((lane >> 4) << 1) + c;
        ws[OFF_FW2P + u] = fW2[n * HID + k];
    } else if (t < 26624) {                            // [pW;tW]: 16 cols, kt(32)
        int u = t - 24576;
        int c = u & 1, lane = (u >> 1) & 31, kt = u >> 6;
        int n = lane & 15;
        int k = kt * 4 + ((lane >> 4) << 1) + c;
        ws[OFF_HEADP + u] = (n < 8) ? pW[n * FEAT + k] : tW[(n - 8) * FEAT + k];
    } else if (t < 92160) {                            // oW1: [8,64,128] -> o,nt(4),kt(32)
        int u = t - 26624;
        int c = u & 1, lane = (u >> 1) & 31, rest = u >> 6;
        int kt = rest & 31, nt = (rest >> 5) & 3, o = rest >> 7;
        int n = nt * 16 + (lane & 15);
        int k = kt * 4 + ((lane >> 4) << 1) + c;
        ws[OFF_OW1P + u] = oW1[(o * HID + n) * FEAT + k];
    } else {                                           // oW2: [8,18,64] pad N->32
        int u = t - 92160;
        int c = u & 1, lane = (u >> 1) & 31, rest = u >> 6;
        int kt = rest & 15, nt = (rest >> 4) & 1, o = rest >> 5;
        int n = nt * 16 + (lane & 15);
        int k = kt * 4 + ((lane >> 4) << 1) + c;
        ws[OFF_OW2P + u] = (n < ACTN) ? oW2[(o * ACTN + n) * HID + k] : 0.f;
    }
}

// ------------------------------------------------------------------
// Option binning: count -> scan -> scatter (deterministic final output:
// each sample writes only its own out[m])
// ------------------------------------------------------------------
__global__ void zero_ints_kernel(int* __restrict__ ints) {
    if (threadIdx.x < 32) ints[threadIdx.x] = 0;
}
__global__ void count_kernel(const int* __restrict__ opt, int* __restrict__ ints) {
    int i = blockIdx.x * blockDim.x + threadIdx.x;
    if (i < BATCH) atomicAdd(&ints[opt[i]], 1);
}
__global__ void scan_kernel(int* __restrict__ ints) {
    if (threadIdx.x == 0) {
        int base = 0;
        for (int o = 0; o < NOPT; ++o) { ints[8 + o] = base; base += ints[o]; }
    }
}
__global__ void scatter_kernel(const int* __restrict__ opt, int* __restrict__ ints,
                               int* __restrict__ idxbuf) {
    int i = blockIdx.x * blockDim.x + threadIdx.x;
    if (i < BATCH) {
        int o = opt[i];
        int pos = atomicAdd(&ints[16 + o], 1);
        idxbuf[ints[8 + o] + pos] = i;
    }
}

// ------------------------------------------------------------------
// Trunk: state = relu(relu(obs@fW1^T+b1)@fW2^T+b2); heads + gather.
// 4 waves/WG, 16 rows/wave, wave-private LDS slices.
// ------------------------------------------------------------------
__global__ __launch_bounds__(128) void trunk_kernel(
    const float* __restrict__ obs, const int* __restrict__ opt,
    const float* __restrict__ fb1, const float* __restrict__ fb2,
    const float* __restrict__ pb, const float* __restrict__ tb,
    const float* __restrict__ fW1p, const float* __restrict__ fW2p,
    const float* __restrict__ headp,
    float* __restrict__ state, float* __restrict__ out) {
    __shared__ float hlds[4][16 * 64];
    __shared__ float slds[4][16 * 128];
    __shared__ float htl[4][16 * 16];

    const int lane = threadIdx.x & 31;
    const int wv = threadIdx.x >> 5;
    const int lh = lane & 15;   // column / row-in-half
    const int hf = lane >> 4;   // half select
    const int m0 = blockIdx.x * 64 + wv * 16;

    // ---- layer 1: [16 x 256] @ [256 x 64]
    v8f acc[4] = {};
    const float* obsRow = obs + (size_t)(m0 + lh) * OBSD + 2 * hf;
    for (int kt = 0; kt < 64; ++kt) {
        v2f a = *(const v2f*)(obsRow + 4 * kt);
#pragma unroll
        for (int nt = 0; nt < 4; ++nt) {
            v2f b = *(const v2f*)(fW1p + ((nt * 64 + kt) * 32 + lane) * 2);
            acc[nt] = wmma4(a, b, acc[nt]);
        }
    }
#pragma unroll
    for (int nt = 0; nt < 4; ++nt) {
        float bias = fb1[nt * 16 + lh];
#pragma unroll
        for (int r = 0; r < 8; ++r) {
            float v = acc[nt][r] + bias;
            hlds[wv][(r + 8 * hf) * 64 + nt * 16 + lh] = v > 0.f ? v : 0.f;
        }
    }
    lds_fence();

    // ---- layer 2: [16 x 64] @ [64 x 128]
    v8f acc2[8] = {};
    for (int kt = 0; kt < 16; ++kt) {
        v2f a = *(const v2f*)&hlds[wv][lh * 64 + 4 * kt + 2 * hf];
#pragma unroll
        for (int nt = 0; nt < 8; ++nt) {
            v2f b = *(const v2f*)(fW2p + ((nt * 16 + kt) * 32 + lane) * 2);
            acc2[nt] = wmma4(a, b, acc2[nt]);
        }
    }
#pragma unroll
    for (int nt = 0; nt < 8; ++nt) {
        float bias = fb2[nt * 16 + lh];
#pragma unroll
        for (int r = 0; r < 8; ++r) {
            float v = acc2[nt][r] + bias;
            v = v > 0.f ? v : 0.f;
            int row = m0 + r + 8 * hf;
            int col = nt * 16 + lh;
            state[(size_t)row * FEAT + col] = v;
            slds[wv][(r + 8 * hf) * 128 + col] = v;
        }
    }
    lds_fence();

    // ---- heads: [16 x 128] @ [128 x 16]  (cols 0-7 policy, 8-15 termination)
    v8f acc3 = {};
    for (int kt = 0; kt < 32; ++kt) {
        v2f a = *(const v2f*)&slds[wv][lh * 128 + 4 * kt + 2 * hf];
        v2f b = *(const v2f*)(headp + (kt * 32 + lane) * 2);
        acc3 = wmma4(a, b, acc3);
    }
    {
        float hb = (lh < 8) ? pb[lh] : tb[lh - 8];
#pragma unroll
        for (int r = 0; r < 8; ++r)
            htl[wv][(r + 8 * hf) * 16 + lh] = acc3[r] + hb;
    }
    lds_fence();
    if (lane < 16) {
        int m = m0 + lane;
        int o = opt[m];
        float ov = htl[wv][lane * 16 + o];
        float tv = htl[wv][lane * 16 + 8 + o];
        out[BATCH + m] = ov;
        out[2 * BATCH + m] = 1.f / (1.f + __expf(-tv));
    }
}

// ------------------------------------------------------------------
// Expert: per (option, 32-row tile), 2 waves/WG, 16 rows/wave.
// logits = relu(state@W1^T+b1)@W2^T+b2 ; out = log_softmax(logits)[act]
// ------------------------------------------------------------------
__global__ __launch_bounds__(64) void expert_kernel(
    const float* __restrict__ state, const int* __restrict__ act,
    const float* __restrict__ ob1, const float* __restrict__ ob2,
    const float* __restrict__ oW1p, const float* __restrict__ oW2p,
    const int* __restrict__ ints, const int* __restrict__ idxbuf,
    float* __restrict__ out) {
    const int o = blockIdx.y;
    const int nrows = ints[o];
    const int tile0 = blockIdx.x * 32;
    if (tile0 >= nrows || nrows == 0) return;

    __shared__ float slds[2][16 * 128];
    __shared__ float ehl[2][16 * 64];
    __shared__ float lgl[2][16 * 32];

    const int lane = threadIdx.x & 31;
    const int wv = threadIdx.x >> 5;
    const int lh = lane & 15;
    const int hf = lane >> 4;
    const int base = ints[8 + o];
    const int row0 = tile0 + wv * 16;

    // gather 16 state rows into LDS (cooperative within wave)
    for (int j = 0; j < 16; ++j) {
        int gr = row0 + j;
        gr = gr < nrows ? gr : nrows - 1;
        int m = idxbuf[base + gr];
        v4f v = *(const v4f*)(state + (size_t)m * FEAT + lane * 4);
        *(v4f*)&slds[wv][j * 128 + lane * 4] = v;
    }
    lds_fence();

    // expert layer 1: [16 x 128] @ [128 x 64]
    v8f acc[4] = {};
    for (int kt = 0; kt < 32; ++kt) {
        v2f a = *(const v2f*)&slds[wv][lh * 128 + 4 * kt + 2 * hf];
#pragma unroll
        for (int nt = 0; nt < 4; ++nt) {
            v2f b = *(const v2f*)(oW1p + o * 8192 + ((nt * 32 + kt) * 32 + lane) * 2);
            acc[nt] = wmma4(a, b, acc[nt]);
        }
    }
#pragma unroll
    for (int nt = 0; nt < 4; ++nt) {
        float bias = ob1[o * HID + nt * 16 + lh];
#pragma unroll
        for (int r = 0; r < 8; ++r) {
            float v = acc[nt][r] + bias;
            ehl[wv][(r + 8 * hf) * 64 + nt * 16 + lh] = v > 0.f ? v : 0.f;
        }
    }
    lds_fence();

    // expert layer 2: [16 x 64] @ [64 x 32(pad)]
    v8f acc2[2] = {};
    for (int kt = 0; kt < 16; ++kt) {
        v2f a = *(const v2f*)&ehl[wv][lh * 64 + 4 * kt + 2 * hf];
#pragma unroll
        for (int nt = 0; nt < 2; ++nt) {
            v2f b = *(const v2f*)(oW2p + o * 2048 + ((nt * 16 + kt) * 32 + lane) * 2);
            acc2[nt] = wmma4(a, b, acc2[nt]);
        }
    }
#pragma unroll
    for (int nt = 0; nt < 2; ++nt) {
        int n = nt * 16 + lh;
        float bias = (n < ACTN) ? ob2[o * ACTN + n] : 0.f;
#pragma unroll
        for (int r = 0; r < 8; ++r)
            lgl[wv][(r + 8 * hf) * 32 + n] = acc2[nt][r] + bias;
    }
    lds_fence();

    // log-softmax + action gather (one lane per sample)
    if (lane < 16) {
        int gr = row0 + lane;
        if (gr < nrows) {
            int m = idxbuf[base + gr];
            const float* lg = &lgl[wv][lane * 32];
            float mx = lg[0];
            for (int t = 1; t < ACTN; ++t) mx = fmaxf(mx, lg[t]);
            float s = 0.f;
            for (int t = 0; t < ACTN; ++t) s += __expf(lg[t] - mx);
            int a = act[m];
            out[m] = lg[a] - mx - __logf(s);
        }
    }
}

extern "C" void kernel_launch(void* const* d_in, const int* in_sizes, int n_in,
                              void* d_out, int out_size, void* d_ws, size_t ws_size,
                              hipStream_t stream) {
    (void)in_sizes; (void)n_in; (void)out_size; (void)ws_size;
    const float* obs = (const float*)d_in[0];
    const int*   act = (const int*)d_in[1];
    const int*   opt = (const int*)d_in[2];
    const float* fW1 = (const float*)d_in[3];
    const float* fb1 = (const float*)d_in[4];
    const float* fW2 = (const float*)d_in[5];
    const float* fb2 = (const float*)d_in[6];
    const float* pW  = (const float*)d_in[7];
    const float* pb  = (const float*)d_in[8];
    const float* tW  = (const float*)d_in[9];
    const float* tb  = (const float*)d_in[10];
    const float* oW1 = (const float*)d_in[11];
    const float* ob1 = (const float*)d_in[12];
    const float* oW2 = (const float*)d_in[13];
    const float* ob2 = (const float*)d_in[14];

    float* ws    = (float*)d_ws;
    float* state = ws + OFF_STATE;
    float* fW1p  = ws + OFF_FW1P;
    float* fW2p  = ws + OFF_FW2P;
    float* headp = ws + OFF_HEADP;
    float* oW1p  = ws + OFF_OW1P;
    float* oW2p  = ws + OFF_OW2P;
    int*   ints  = (int*)(ws + OFF_INTS);
    int*   idxb  = (int*)(ws + OFF_IDX);
    float* out   = (float*)d_out;

    pack_weights_kernel<<<(108544 + 255) / 256, 256, 0, stream>>>(
        fW1, fW2, pW, tW, oW1, oW2, ws);
    zero_ints_kernel<<<1, 32, 0, stream>>>(ints);
    count_kernel<<<BATCH / 256, 256, 0, stream>>>(opt, ints);
    scan_kernel<<<1, 32, 0, stream>>>(ints);
    scatter_kernel<<<BATCH / 256, 256, 0, stream>>>(opt, ints, idxb);

    trunk_kernel<<<BATCH / 64, 128, 0, stream>>>(
        obs, opt, fb1, fb2, pb, tb, fW1p, fW2p, headp, state, out);

    dim3 egrid(BATCH / 32, NOPT);
    expert_kernel<<<egrid, 64, 0, stream>>>(
        state, act, ob1, ob2, oW1p, oW2p, ints, idxb, out);
}